// ROIPooling_11287174054520
// MI455X (gfx1250) — compile-verified
//
#include <hip/hip_runtime.h>
#include <cstdint>
#include <math.h>

// ROI pooling for MI455X (gfx1250).
// Memory-bound gather op: ~25.7MB out + ~10.2MB unique in (input lives in L2).
// No GEMM structure -> WMMA not applicable; use wave32 float4 coalesced gathers
// + async global->LDS staging (ASYNCcnt) to pipeline the 4 corner fetches.

#define AS1 __attribute__((address_space(1)))
#define AS3 __attribute__((address_space(3)))

namespace {
constexpr int FH     = 50;
constexpr int FW     = 50;
constexpr int FC     = 512;
constexpr int POOLED = 7;
constexpr int CROP   = 14;
constexpr int TPB    = 128;   // 4 waves; each thread owns channels [4*tid, 4*tid+3]
constexpr int NBUF   = 4;     // LDS staging buffers (deep reuse distance, no WAR hazard)
}

#if defined(__gfx1250__) &&                                                   \
    __has_builtin(__builtin_amdgcn_global_load_async_to_lds_b128) &&          \
    __has_builtin(__builtin_amdgcn_s_wait_asynccnt)
#define ROI_ASYNC 1
#endif

#ifndef ROI_ASYNC
#define ROI_ASYNC 0
#endif

#if ROI_ASYNC
typedef int v4i __attribute__((ext_vector_type(4)));
#endif

__device__ __forceinline__ float4 f4lerp(float4 a, float4 b, float w) {
    return make_float4(a.x + (b.x - a.x) * w,
                       a.y + (b.y - a.y) * w,
                       a.z + (b.z - a.z) * w,
                       a.w + (b.w - a.w) * w);
}
__device__ __forceinline__ float4 f4max(float4 a, float4 b) {
    return make_float4(fmaxf(a.x, b.x), fmaxf(a.y, b.y),
                       fmaxf(a.z, b.z), fmaxf(a.w, b.w));
}

struct XSamp { int x0; int x1; float wx; bool vx; };

__device__ __forceinline__ XSamp xsample(float bx, float sx, int j) {
    XSamp s;
    const float xs = bx + (float)j * sx;
    s.vx = (xs >= 0.0f) && (xs <= (float)(FW - 1));
    const float xf = floorf(xs);
    s.wx = xs - xf;
    int x0 = (int)xf;
    x0 = x0 < 0 ? 0 : (x0 > FW - 1 ? FW - 1 : x0);
    s.x0 = x0;
    s.x1 = (x0 + 1 > FW - 1) ? FW - 1 : x0 + 1;
    return s;
}

__global__ __launch_bounds__(TPB) void roi_pool_kernel(
    const float* __restrict__ fmap,
    const float* __restrict__ rois,
    float* __restrict__ out)
{
    const int n   = blockIdx.x;    // 0..255 : batch*128 + roi
    const int py  = blockIdx.y;    // 0..6   : pooled row
    const int tid = threadIdx.x;   // channel quad
    const int b   = n >> 7;

    // rois layout (x1, y1, x2, y2) after the reference's [1,0,3,2] permute:
    // y1 = rois[.,1], x1 = rois[.,0], y2 = rois[.,3], x2 = rois[.,2]
    const float4 r = reinterpret_cast<const float4*>(rois)[n];
    const float rx1 = r.x, ry1 = r.y, rx2 = r.z, ry2 = r.w;

    const float by = ry1 * (float)(FH - 1);
    const float bx = rx1 * (float)(FW - 1);
    const float sy = (ry2 - ry1) * (float)(FH - 1) / (float)(CROP - 1);
    const float sx = (rx2 - rx1) * (float)(FW - 1) / (float)(CROP - 1);

    const float* bbase = fmap + (size_t)b * (FH * FW * FC);

    float4 acc[POOLED];

#if ROI_ASYNC
    __shared__ float sbuf[NBUF][4][TPB * 4];   // NBUF x 4 corners x 2KB = 32KB
#endif

#pragma unroll
    for (int ky = 0; ky < 2; ++ky) {
        const int   i   = 2 * py + ky;
        const float ys  = by + (float)i * sy;
        const bool  vyb = (ys >= 0.0f) && (ys <= (float)(FH - 1));
        const float yf  = floorf(ys);
        const float wy  = ys - yf;
        int y0 = (int)yf;
        y0 = y0 < 0 ? 0 : (y0 > FH - 1 ? FH - 1 : y0);
        const int y1 = (y0 + 1 > FH - 1) ? FH - 1 : y0 + 1;
        const float* row0 = bbase + (size_t)y0 * (FW * FC);
        const float* row1 = bbase + (size_t)y1 * (FW * FC);

        auto fold = [&](int j, float4 v) {
            const int px = j >> 1;
            if (ky == 0 && (j & 1) == 0) acc[px] = v;
            else                         acc[px] = f4max(acc[px], v);
        };

#if ROI_ASYNC
        // Pipeline: stage the 4 corner float4s for sample j+1 into LDS via
        // async loads while consuming sample j. Each lane reads back exactly
        // the slot it filled -> no workgroup barrier, only s_wait_asynccnt.
        auto issue = [&](int j, int buf) {
            const XSamp s = xsample(bx, sx, j);
            const float* g00 = row0 + (size_t)s.x0 * FC + (tid << 2);
            const float* g01 = row0 + (size_t)s.x1 * FC + (tid << 2);
            const float* g10 = row1 + (size_t)s.x0 * FC + (tid << 2);
            const float* g11 = row1 + (size_t)s.x1 * FC + (tid << 2);
            __builtin_amdgcn_global_load_async_to_lds_b128(
                (AS1 v4i*)(uintptr_t)g00, (AS3 v4i*)&sbuf[buf][0][tid << 2], 0, 0);
            __builtin_amdgcn_global_load_async_to_lds_b128(
                (AS1 v4i*)(uintptr_t)g01, (AS3 v4i*)&sbuf[buf][1][tid << 2], 0, 0);
            __builtin_amdgcn_global_load_async_to_lds_b128(
                (AS1 v4i*)(uintptr_t)g10, (AS3 v4i*)&sbuf[buf][2][tid << 2], 0, 0);
            __builtin_amdgcn_global_load_async_to_lds_b128(
                (AS1 v4i*)(uintptr_t)g11, (AS3 v4i*)&sbuf[buf][3][tid << 2], 0, 0);
        };
        auto consume = [&](int j, int buf) {
            const XSamp s = xsample(bx, sx, j);
            const float4 a00 = *reinterpret_cast<const float4*>(&sbuf[buf][0][tid << 2]);
            const float4 a01 = *reinterpret_cast<const float4*>(&sbuf[buf][1][tid << 2]);
            const float4 a10 = *reinterpret_cast<const float4*>(&sbuf[buf][2][tid << 2]);
            const float4 a11 = *reinterpret_cast<const float4*>(&sbuf[buf][3][tid << 2]);
            float4 v = f4lerp(f4lerp(a00, a10, wy), f4lerp(a01, a11, wy), s.wx);
            if (!(vyb && s.vx)) v = make_float4(0.f, 0.f, 0.f, 0.f);
            fold(j, v);
        };

        issue(0, 0);
#pragma unroll
        for (int j = 0; j < CROP - 1; ++j) {
            issue(j + 1, (j + 1) & (NBUF - 1));
            __builtin_amdgcn_s_wait_asynccnt(4);   // 4 newest may be in flight; j's 4 are done
            asm volatile("" ::: "memory");
            consume(j, j & (NBUF - 1));
        }
        __builtin_amdgcn_s_wait_asynccnt(0);
        asm volatile("" ::: "memory");
        consume(CROP - 1, (CROP - 1) & (NBUF - 1));
#else
        // Fallback: direct coalesced b128 gathers (latency hidden by unroll + waves)
#pragma unroll
        for (int j = 0; j < CROP; ++j) {
            const XSamp s = xsample(bx, sx, j);
            const float4 a00 = reinterpret_cast<const float4*>(row0 + (size_t)s.x0 * FC)[tid];
            const float4 a01 = reinterpret_cast<const float4*>(row0 + (size_t)s.x1 * FC)[tid];
            const float4 a10 = reinterpret_cast<const float4*>(row1 + (size_t)s.x0 * FC)[tid];
            const float4 a11 = reinterpret_cast<const float4*>(row1 + (size_t)s.x1 * FC)[tid];
            float4 v = f4lerp(f4lerp(a00, a10, wy), f4lerp(a01, a11, wy), s.wx);
            if (!(vyb && s.vx)) v = make_float4(0.f, 0.f, 0.f, 0.f);
            fold(j, v);
        }
#endif
    }

    float* obase = out + (((size_t)n * POOLED + py) * POOLED) * FC;
#pragma unroll
    for (int px = 0; px < POOLED; ++px) {
        reinterpret_cast<float4*>(obase + (size_t)px * FC)[tid] = acc[px];
    }
}

extern "C" void kernel_launch(void* const* d_in, const int* in_sizes, int n_in,
                              void* d_out, int out_size, void* d_ws, size_t ws_size,
                              hipStream_t stream) {
    (void)n_in; (void)out_size; (void)d_ws; (void)ws_size;
    const float* fmap = (const float*)d_in[0];
    const float* rois = (const float*)d_in[1];
    float* out = (float*)d_out;

    const int n_rois = in_sizes[1] / 4;       // 256
    dim3 grid(n_rois, POOLED);
    roi_pool_kernel<<<grid, TPB, 0, stream>>>(fmap, rois, out);
}